// MatchingLayer_48241072668683
// MI455X (gfx1250) — compile-verified
//
#include <hip/hip_runtime.h>
#include <math.h>

typedef __attribute__((ext_vector_type(2))) float v2f;
typedef __attribute__((ext_vector_type(8))) float v8f;

#define NB   16      // batch
#define LL   128     // seq len
#define HD   100     // hidden
#define PD   20      // perspectives
#define CH   210     // output channels per side
#define C2H  200     // 2*HD row stride
#define EPSF 1e-8f

// ---------------- workspace layout (float offsets) ----------------
#define OFF_COS      0                                   // [dir][b][i][j]
#define SZ_COS       (2*NB*LL*LL)
#define OFF_ROWSUM   (OFF_COS + SZ_COS)                  // [dir][b][i]
#define OFF_ROWMAX   (OFF_ROWSUM + 2*NB*LL)
#define OFF_COLSUM   (OFF_ROWMAX + 2*NB*LL)
#define OFF_COLMAX   (OFF_COLSUM + 2*NB*LL)
#define OFF_NPFW     (OFF_COLMAX + 2*NB*LL)              // [b][i]
#define OFF_NPBW     (OFF_NPFW + NB*LL)
#define OFF_NHFW     (OFF_NPBW + NB*LL)
#define OFF_NHBW     (OFF_NHFW + NB*LL)
#define OFF_WNP      (OFF_NHBW + NB*LL)                  // [dir][b][p][i]
#define OFF_WNH      (OFF_WNP + 2*NB*PD*LL)
#define OFF_AMEAN_H  (OFF_WNH + 2*NB*PD*LL)              // [dir][b][i][h]
#define OFF_AMAX_H   (OFF_AMEAN_H + 2*NB*LL*HD)
#define OFF_AMEAN_P  (OFF_AMAX_H + 2*NB*LL*HD)
#define OFF_AMAX_P   (OFF_AMEAN_P + 2*NB*LL*HD)
#define OFF_MROWMAX  (OFF_AMAX_P + 2*NB*LL*HD)           // [dir][b][i][p]
#define OFF_MROWMEAN (OFF_MROWMAX + 2*NB*LL*PD)
#define OFF_MCOLMAX  (OFF_MROWMEAN + 2*NB*LL*PD)
#define OFF_MCOLMEAN (OFF_MCOLMAX + 2*NB*LL*PD)
#define OFF_LAST     (OFF_MCOLMEAN + 2*NB*LL*PD)         // 2*NB ints (lastP, lastH)

__device__ __forceinline__ float negInf() { return -__builtin_inff(); }

// ---------------- K0: plain norms + last indices ----------------
__global__ void k0_norms(const float* __restrict__ cp, const float* __restrict__ ch,
                         const int* __restrict__ mp, const int* __restrict__ mh,
                         float* __restrict__ ws) {
    int b = blockIdx.x;
    int i = threadIdx.x;            // 0..127
    const float* pr = cp + (size_t)(b*LL + i)*C2H;
    const float* hr = ch + (size_t)(b*LL + i)*C2H;
    float s0=0.f, s1=0.f, s2=0.f, s3=0.f;
    for (int h = 0; h < HD; ++h) {
        float a = pr[h], bb = pr[HD+h], c = hr[h], d = hr[HD+h];
        s0 += a*a; s1 += bb*bb; s2 += c*c; s3 += d*d;
    }
    ws[OFF_NPFW + b*LL + i] = sqrtf(s0);
    ws[OFF_NPBW + b*LL + i] = sqrtf(s1);
    ws[OFF_NHFW + b*LL + i] = sqrtf(s2);
    ws[OFF_NHBW + b*LL + i] = sqrtf(s3);
    if (i == 0) {
        int sp = 0, sh = 0;
        for (int k = 0; k < LL; ++k) { sp += mp[b*LL+k]; sh += mh[b*LL+k]; }
        int lp = sp - 1; if (lp < 0) lp = 0;
        int lh = sh - 1; if (lh < 0) lh = 0;
        ((int*)(ws + OFF_LAST))[b]      = lp;
        ((int*)(ws + OFF_LAST))[NB + b] = lh;
    }
}

// ---------------- K0b: weighted norms for mpm_pairwise ----------------
__global__ void k0_wnorms(const float* __restrict__ cp, const float* __restrict__ ch,
                          const float* __restrict__ params, float* __restrict__ ws) {
    int b = blockIdx.x, p = blockIdx.y;
    int i = threadIdx.x;            // 0..127
    __shared__ float w2f[HD], w2b[HD];
    if (i < HD) {
        float wf = params[(2*PD + p)*HD + i];
        float wb = params[(3*PD + p)*HD + i];
        w2f[i] = wf*wf; w2b[i] = wb*wb;
    }
    __syncthreads();
    const float* pr = cp + (size_t)(b*LL + i)*C2H;
    const float* hr = ch + (size_t)(b*LL + i)*C2H;
    float s0=0.f, s1=0.f, s2=0.f, s3=0.f;
    for (int h = 0; h < HD; ++h) {
        float pf = pr[h], hf = hr[h], pb = pr[HD+h], hb = hr[HD+h];
        s0 += w2f[h]*pf*pf; s1 += w2f[h]*hf*hf;
        s2 += w2b[h]*pb*pb; s3 += w2b[h]*hb*hb;
    }
    ws[OFF_WNP + (size_t)((0*NB + b)*PD + p)*LL + i] = sqrtf(s0);
    ws[OFF_WNH + (size_t)((0*NB + b)*PD + p)*LL + i] = sqrtf(s1);
    ws[OFF_WNP + (size_t)((1*NB + b)*PD + p)*LL + i] = sqrtf(s2);
    ws[OFF_WNH + (size_t)((1*NB + b)*PD + p)*LL + i] = sqrtf(s3);
}

// ---------------- K1: cosine_pairwise via WMMA f32 16x16x4 ----------------
__global__ void __launch_bounds__(32) k1_cos(const float* __restrict__ cp,
                                             const float* __restrict__ ch,
                                             float* __restrict__ ws) {
    int blk  = blockIdx.x;          // NB*8*2
    int dir  = blk & 1;
    int tI   = (blk >> 1) & 7;
    int b    = blk >> 4;
    int lane = threadIdx.x;
    int half = lane >> 4;
    int lm   = lane & 15;
    int off  = dir * HD;

    // Preload A tile (16 rows x 100 cols) into 25 float2 per lane.
    v2f a[25];
    int iRow = tI*16 + lm;
    const float* ar = cp + (size_t)(b*LL + iRow)*C2H + off;
#pragma unroll
    for (int kk = 0; kk < 25; ++kk)
        a[kk] = *(const v2f*)(ar + kk*4 + 2*half);

    const float* n1base = ws + (dir ? OFF_NPBW : OFF_NPFW) + b*LL;
    const float* n2base = ws + (dir ? OFF_NHBW : OFF_NHFW) + b*LL;
    float* cosOut = ws + OFF_COS + (size_t)((dir*NB + b)*LL)*LL;

    for (int tJ = 0; tJ < 8; ++tJ) {
        int j = tJ*16 + lm;
        const float* br = ch + (size_t)(b*LL + j)*C2H + off;
        // Stage the whole B tile in registers first: one load clause, one wait,
        // then 25 back-to-back WMMAs on the matrix pipe.
        v2f bt[25];
#pragma unroll
        for (int kk = 0; kk < 25; ++kk)
            bt[kk] = *(const v2f*)(br + kk*4 + 2*half);
        v8f c = {};
#pragma unroll
        for (int kk = 0; kk < 25; ++kk)
            c = __builtin_amdgcn_wmma_f32_16x16x4_f32(false, a[kk], false, bt[kk],
                                                      (short)0, c, false, false);
        float n2 = n2base[j];
#pragma unroll
        for (int r = 0; r < 8; ++r) {
            int i = tI*16 + half*8 + r;
            float den = fmaxf(n1base[i] * n2, EPSF);   // div_safe
            cosOut[(size_t)i*LL + j] = c[r] / den;
        }
    }
}

// ---------------- K2: row/col max & sum of cos matrices ----------------
__global__ void k2_stats(float* __restrict__ ws) {
    int dir = blockIdx.x & 1, b = blockIdx.x >> 1;
    int t = threadIdx.x;            // 0..127
    const float* cm = ws + OFF_COS + (size_t)((dir*NB + b)*LL)*LL;
    float rs = 0.f, rm = negInf(), cs = 0.f, cmx = negInf();
    for (int k = 0; k < LL; ++k) {
        float rv = cm[(size_t)t*LL + k];
        rs += rv; rm = fmaxf(rm, rv);
        float cv = cm[(size_t)k*LL + t];
        cs += cv; cmx = fmaxf(cmx, cv);
    }
    int idx = (dir*NB + b)*LL + t;
    ws[OFF_ROWSUM + idx] = rs;  ws[OFF_ROWMAX + idx] = rm;
    ws[OFF_COLSUM + idx] = cs;  ws[OFF_COLMAX + idx] = cmx;
}

// ---------------- K3: attentive mean / max vectors ----------------
__global__ void k3_att(const float* __restrict__ cp, const float* __restrict__ ch,
                       float* __restrict__ ws) {
    int combo = blockIdx.x & 3;     // 0:h_fw 1:h_bw 2:p_fw 3:p_bw
    int b     = blockIdx.x >> 2;
    int dir   = combo & 1;
    int pside = combo >> 1;         // 0: reduce over j with h vecs, 1: over i with p vecs
    int t = threadIdx.x;            // 0..127
    __shared__ float V[LL*HD];      // 51.2 KB
    __shared__ float crow[LL];
    const float* src = pside ? cp : ch;
    int off = dir*HD;
    for (int idx = t; idx < LL*HD; idx += blockDim.x) {
        int k = idx / HD, h = idx % HD;
        V[idx] = src[(size_t)(b*LL + k)*C2H + off + h];
    }
    const float* cm   = ws + OFF_COS + (size_t)((dir*NB + b)*LL)*LL;
    const float* den  = ws + (pside ? OFF_COLSUM : OFF_ROWSUM) + (dir*NB + b)*LL;
    float* outMean = ws + (pside ? OFF_AMEAN_P : OFF_AMEAN_H) + (size_t)((dir*NB + b)*LL)*HD;
    float* outMax  = ws + (pside ? OFF_AMAX_P  : OFF_AMAX_H ) + (size_t)((dir*NB + b)*LL)*HD;
    for (int o = 0; o < LL; ++o) {
        __syncthreads();
        if (t < LL) crow[t] = pside ? cm[(size_t)t*LL + o] : cm[(size_t)o*LL + t];
        __syncthreads();
        if (t < HD) {
            float acc = 0.f, mx = negInf();
            for (int k = 0; k < LL; ++k) {
                float v = crow[k] * V[k*HD + t];
                acc += v; mx = fmaxf(mx, v);
            }
            outMean[(size_t)o*HD + t] = acc / fmaxf(den[o], EPSF);
            outMax [(size_t)o*HD + t] = mx;
        }
    }
}

// ---------------- K4: mpm_pairwise via WMMA + in-wave reductions ----------------
__global__ void __launch_bounds__(32) k4_mpm_pair(const float* __restrict__ cp,
                                                  const float* __restrict__ ch,
                                                  const float* __restrict__ params,
                                                  float* __restrict__ ws) {
    int p    = blockIdx.x % PD;
    int rest = blockIdx.x / PD;
    int b    = rest % NB;
    int dir  = rest / NB;
    int lane = threadIdx.x, half = lane >> 4, lm = lane & 15;
    int off  = dir * HD;

    __shared__ float rowS[LL], rowM[LL], colS[LL], colM[LL], wn1s[LL], wn2s[LL];
    const float* wn1g = ws + OFF_WNP + (size_t)((dir*NB + b)*PD + p)*LL;
    const float* wn2g = ws + OFF_WNH + (size_t)((dir*NB + b)*PD + p)*LL;
    for (int k = lane; k < LL; k += 32) {
        rowS[k] = 0.f; rowM[k] = negInf();
        colS[k] = 0.f; colM[k] = negInf();
        wn1s[k] = wn1g[k]; wn2s[k] = wn2g[k];
    }
    __syncthreads();

    // per-lane squared weights at this lane's K positions
    v2f w2[25];
    const float* wrow = params + ((2 + dir)*PD + p)*HD;
#pragma unroll
    for (int kk = 0; kk < 25; ++kk) {
        int k0 = kk*4 + 2*half;
        float wa = wrow[k0], wb = wrow[k0+1];
        v2f t; t.x = wa*wa; t.y = wb*wb;
        w2[kk] = t;
    }

    for (int tI = 0; tI < 8; ++tI) {
        int i0 = tI*16;
        const float* ar = cp + (size_t)(b*LL + i0 + lm)*C2H + off;
        v2f a[25];
#pragma unroll
        for (int kk = 0; kk < 25; ++kk) {
            v2f av = *(const v2f*)(ar + kk*4 + 2*half);
            a[kk] = av * w2[kk];            // fold w^2 into A operand
        }
        for (int tJ = 0; tJ < 8; ++tJ) {
            int j = tJ*16 + lm;
            const float* br = ch + (size_t)(b*LL + j)*C2H + off;
            // Stage whole B tile, then dense WMMA chain.
            v2f bt[25];
#pragma unroll
            for (int kk = 0; kk < 25; ++kk)
                bt[kk] = *(const v2f*)(br + kk*4 + 2*half);
            v8f c = {};
#pragma unroll
            for (int kk = 0; kk < 25; ++kk)
                c = __builtin_amdgcn_wmma_f32_16x16x4_f32(false, a[kk], false, bt[kk],
                                                          (short)0, c, false, false);
            float wn2 = wn2s[j];
            float mv[8];
#pragma unroll
            for (int r = 0; r < 8; ++r) {
                int i = i0 + half*8 + r;
                mv[r] = c[r] / fmaxf(wn1s[i]*wn2, EPSF);   // div_safe
            }
            // row reduction over j (across each 16-lane half)
#pragma unroll
            for (int r = 0; r < 8; ++r) {
                float s = mv[r], m = mv[r];
                for (int d = 1; d < 16; d <<= 1) {
                    s += __shfl_xor(s, d, 32);
                    m  = fmaxf(m, __shfl_xor(m, d, 32));
                }
                if (lm == 0) {
                    int i = i0 + half*8 + r;
                    rowS[i] += s;
                    rowM[i]  = fmaxf(rowM[i], m);
                }
            }
            // column reduction over i (in-lane over r, then across halves)
            float cs = 0.f, cmx = negInf();
#pragma unroll
            for (int r = 0; r < 8; ++r) { cs += mv[r]; cmx = fmaxf(cmx, mv[r]); }
            cs  += __shfl_xor(cs, 16, 32);
            cmx  = fmaxf(cmx, __shfl_xor(cmx, 16, 32));
            if (half == 0) {
                colS[j] += cs;
                colM[j]  = fmaxf(colM[j], cmx);
            }
        }
    }
    __syncthreads();
    float* rmax  = ws + OFF_MROWMAX  + (size_t)((dir*NB + b)*LL)*PD;
    float* rmean = ws + OFF_MROWMEAN + (size_t)((dir*NB + b)*LL)*PD;
    float* cmax  = ws + OFF_MCOLMAX  + (size_t)((dir*NB + b)*LL)*PD;
    float* cmean = ws + OFF_MCOLMEAN + (size_t)((dir*NB + b)*LL)*PD;
    const float invL = 1.0f / (float)LL;
    for (int k = lane; k < LL; k += 32) {
        rmax [k*PD + p] = rowM[k];
        rmean[k*PD + p] = rowS[k] * invL;
        cmax [k*PD + p] = colM[k];
        cmean[k*PD + p] = colS[k] * invL;
    }
}

// ---------------- K5: assemble 210 channels per (side,b,i) ----------------
__global__ void k5_final(const float* __restrict__ cp, const float* __restrict__ ch,
                         const float* __restrict__ params, float* __restrict__ ws,
                         float* __restrict__ out) {
    int i = blockIdx.x, b = blockIdx.y, side = blockIdx.z;   // side 0: mv_p, 1: mv_h
    int t = threadIdx.x;                                     // 0..127
    __shared__ float v1f[HD], v1b[HD], ofv[HD], obv[HD], amf[HD], amb[HD], axf[HD], axb[HD];
    const int* lastArr = (const int*)(ws + OFF_LAST);
    const float* self  = side ? ch : cp;
    const float* other = side ? cp : ch;
    int lastOther = side ? lastArr[b] : lastArr[NB + b];
    if (t < HD) {
        v1f[t] = self [(size_t)(b*LL + i)*C2H + t];
        v1b[t] = self [(size_t)(b*LL + i)*C2H + HD + t];
        ofv[t] = other[(size_t)(b*LL + lastOther)*C2H + t];
        obv[t] = other[(size_t)(b*LL + 0)*C2H + HD + t];
        const float* mean = ws + (side ? OFF_AMEAN_P : OFF_AMEAN_H);
        const float* mx   = ws + (side ? OFF_AMAX_P  : OFF_AMAX_H );
        amf[t] = mean[(size_t)((0*NB + b)*LL + i)*HD + t];
        amb[t] = mean[(size_t)((1*NB + b)*LL + i)*HD + t];
        axf[t] = mx  [(size_t)((0*NB + b)*LL + i)*HD + t];
        axb[t] = mx  [(size_t)((1*NB + b)*LL + i)*HD + t];
    }
    __syncthreads();
    float* o = out + ((size_t)side*NB*LL + (size_t)b*LL + i)*CH;

    const float* V1[6] = { v1f, v1b, v1f, v1b, v1f, v1b };
    const float* V2[6] = { ofv, obv, amf, amb, axf, axb };
    const int MIDX [6] = { 0, 1, 4, 5, 6, 7 };
    const int CBASE[6] = { 4, 25, 126, 147, 168, 189 };

    if (t < 120) {              // weighted cosine: combo x perspective
        int cb = t / 20, pp = t % 20;
        const float* w  = params + (MIDX[cb]*PD + pp)*HD;
        const float* aa = V1[cb];
        const float* bb = V2[cb];
        float s11 = 0.f, s22 = 0.f, s12 = 0.f;
        for (int h = 0; h < HD; ++h) {
            float ww = w[h]*w[h], av = aa[h], bv = bb[h];
            s11 += ww*av*av; s22 += ww*bv*bv; s12 += ww*av*bv;
        }
        float n1 = fmaxf(sqrtf(s11), EPSF), n2 = fmaxf(sqrtf(s22), EPSF);
        o[CBASE[cb] + 1 + pp] = s12 / (n1*n2);
    } else if (t < 126) {       // unweighted cosine per combo
        int cb = t - 120;
        const float* aa = V1[cb];
        const float* bb = V2[cb];
        float s11 = 0.f, s22 = 0.f, s12 = 0.f;
        for (int h = 0; h < HD; ++h) {
            float av = aa[h], bv = bb[h];
            s11 += av*av; s22 += bv*bv; s12 += av*bv;
        }
        float n1 = fmaxf(sqrtf(s11), EPSF), n2 = fmaxf(sqrtf(s22), EPSF);
        o[CBASE[cb]] = s12 / (n1*n2);
    } else if (t == 126) {      // channels 0..3: cos max/mean
        const float invL = 1.0f / (float)LL;
        int idxF = (0*NB + b)*LL + i, idxB = (1*NB + b)*LL + i;
        if (side == 0) {
            o[0] = ws[OFF_ROWMAX + idxF]; o[1] = ws[OFF_ROWSUM + idxF]*invL;
            o[2] = ws[OFF_ROWMAX + idxB]; o[3] = ws[OFF_ROWSUM + idxB]*invL;
        } else {
            o[0] = ws[OFF_COLMAX + idxF]; o[1] = ws[OFF_COLSUM + idxF]*invL;
            o[2] = ws[OFF_COLMAX + idxB]; o[3] = ws[OFF_COLSUM + idxB]*invL;
        }
    }
    if (t < 80) {               // channels 46..125: mpm_pairwise reductions
        int q = t / 20, pp = t % 20;
        int dir = q >> 1;
        bool isMax = ((q & 1) == 0);
        size_t idx = ((size_t)(dir*NB + b)*LL + i)*PD + pp;
        const float* aMax  = ws + (side ? OFF_MCOLMAX  : OFF_MROWMAX);
        const float* aMean = ws + (side ? OFF_MCOLMEAN : OFF_MROWMEAN);
        o[46 + q*20 + pp] = isMax ? aMax[idx] : aMean[idx];
    }
}

// ---------------- launcher ----------------
extern "C" void kernel_launch(void* const* d_in, const int* in_sizes, int n_in,
                              void* d_out, int out_size, void* d_ws, size_t ws_size,
                              hipStream_t stream) {
    (void)in_sizes; (void)n_in; (void)out_size; (void)ws_size;
    const float* cp     = (const float*)d_in[0];
    const int*   mp     = (const int*)  d_in[1];
    const float* ch     = (const float*)d_in[2];
    const int*   mh     = (const int*)  d_in[3];
    const float* params = (const float*)d_in[4];
    float* ws  = (float*)d_ws;
    float* out = (float*)d_out;

    k0_norms    <<<NB,               LL, 0, stream>>>(cp, ch, mp, mh, ws);
    k0_wnorms   <<<dim3(NB, PD),     LL, 0, stream>>>(cp, ch, params, ws);
    k1_cos      <<<NB*8*2,           32, 0, stream>>>(cp, ch, ws);
    k2_stats    <<<NB*2,             LL, 0, stream>>>(ws);
    k3_att      <<<NB*4,             LL, 0, stream>>>(cp, ch, ws);
    k4_mpm_pair <<<2*NB*PD,          32, 0, stream>>>(cp, ch, params, ws);
    k5_final    <<<dim3(LL, NB, 2),  LL, 0, stream>>>(cp, ch, params, ws, out);
}